// CompoundEventModel_38955353375020
// MI455X (gfx1250) — compile-verified
//
#include <hip/hip_runtime.h>

#define HID 64
#define N_GRAPHS 64

typedef __attribute__((ext_vector_type(2))) float v2f;
typedef __attribute__((ext_vector_type(8))) float v8f;

// ---------------- utility ----------------
__global__ void zero_kernel(float* __restrict__ p, long long n) {
    long long i = (long long)blockIdx.x * blockDim.x + threadIdx.x;
    long long stride = (long long)gridDim.x * blockDim.x;
    for (; i < n; i += stride) p[i] = 0.0f;
}

__global__ void degree_kernel(const int* __restrict__ dst, float* __restrict__ deg, int E) {
    int e = blockIdx.x * blockDim.x + threadIdx.x;
    if (e < E) atomicAdd(&deg[dst[e]], 1.0f);
}

__global__ void deginv_kernel(const float* __restrict__ deg, float* __restrict__ dinv, int N) {
    int n = blockIdx.x * blockDim.x + threadIdx.x;
    if (n < N) {
        float d = deg[n];
        dinv[n] = (d > 0.0f) ? (1.0f / fmaxf(d, 1.0f)) : 0.0f;
    }
}

// ---------------- edge scatter: msg[dst] += feat[src] ----------------
template<int D>
__global__ void scatter_add_kernel(const float* __restrict__ feat,
                                   const int* __restrict__ src,
                                   const int* __restrict__ dst,
                                   float* __restrict__ msg, int E) {
    constexpr int CH = D / 4;  // float4 chunks per edge
    long long t = (long long)blockIdx.x * blockDim.x + threadIdx.x;
    if (t >= (long long)E * CH) return;
    int e  = (int)(t / CH);
    int c4 = (int)(t % CH) * 4;
    int s = src[e], d = dst[e];
    const float4 v = *(const float4*)(feat + (long long)s * D + c4);
    float* mp = msg + (long long)d * D + c4;
    atomicAdd(mp + 0, v.x);
    atomicAdd(mp + 1, v.y);
    atomicAdd(mp + 2, v.z);
    atomicAdd(mp + 3, v.w);
}

// ---------------- fused SAGE node transform via fp32 WMMA ----------------
// out[n,:] = relu?( (msg[n,:]*dinv[n]) @ Wl + bl + hin[n,:] @ Wr )
// One wave computes a 16x64 output tile using V_WMMA_F32_16X16X4_F32.
template<int K, bool RELU>
__global__ __launch_bounds__(128)
void sage_gemm_kernel(const float* __restrict__ msg, const float* __restrict__ dinv,
                      const float* __restrict__ hin,
                      const float* __restrict__ Wl, const float* __restrict__ bl,
                      const float* __restrict__ Wr,
                      float* __restrict__ hout, int ntiles) {
    __shared__ float sW[2][K * HID];
    int tid = threadIdx.x;
    for (int i = tid; i < K * HID; i += 128) {
        sW[0][i] = Wl[i];
        sW[1][i] = Wr[i];
    }
    __syncthreads();

    int wave = tid >> 5;
    int lane = tid & 31;
    int tile = blockIdx.x * 4 + wave;
    if (tile >= ntiles) return;           // wave-uniform; EXEC all-1s for WMMA
    int row0 = tile << 4;
    int half = lane >> 4;                 // 0: K pair {4k,4k+1}, 1: {4k+2,4k+3}
    int l15  = lane & 15;
    int arow = row0 + l15;
    float scale = dinv[arow];

    v8f acc[4] = {};

    const float* aM = msg + (long long)arow * K + 2 * half;
    const float* aH = hin + (long long)arow * K + 2 * half;

    // path 1: mean-neighbor features @ Wl
#pragma unroll
    for (int k = 0; k < K / 4; ++k) {
        v2f a;
        a.x = aM[4 * k + 0] * scale;
        a.y = aM[4 * k + 1] * scale;
        int krow = 4 * k + 2 * half;
#pragma unroll
        for (int c = 0; c < 4; ++c) {
            v2f b;
            b.x = sW[0][(krow + 0) * HID + c * 16 + l15];
            b.y = sW[0][(krow + 1) * HID + c * 16 + l15];
            acc[c] = __builtin_amdgcn_wmma_f32_16x16x4_f32(
                false, a, false, b, (short)0, acc[c], false, false);
        }
    }
    // path 2: self features @ Wr (accumulate into same C)
#pragma unroll
    for (int k = 0; k < K / 4; ++k) {
        v2f a;
        a.x = aH[4 * k + 0];
        a.y = aH[4 * k + 1];
        int krow = 4 * k + 2 * half;
#pragma unroll
        for (int c = 0; c < 4; ++c) {
            v2f b;
            b.x = sW[1][(krow + 0) * HID + c * 16 + l15];
            b.y = sW[1][(krow + 1) * HID + c * 16 + l15];
            acc[c] = __builtin_amdgcn_wmma_f32_16x16x4_f32(
                false, a, false, b, (short)0, acc[c], false, false);
        }
    }

    // D layout: VGPR v -> row (v + 8*half), N = l15 within each 16-col tile
#pragma unroll
    for (int c = 0; c < 4; ++c) {
        int col = c * 16 + l15;
        float bias = bl[col];
#pragma unroll
        for (int v = 0; v < 8; ++v) {
            float val = acc[c][v] + bias;
            if (RELU) val = fmaxf(val, 0.0f);
            hout[(long long)(row0 + v + 8 * half) * HID + col] = val;
        }
    }
}

// ---------------- per-graph mean pool (LDS-staged atomics) ----------------
__global__ __launch_bounds__(256)
void pool_kernel(const float* __restrict__ h, const int* __restrict__ batch,
                 float* __restrict__ pooled, float* __restrict__ counts, int N) {
    __shared__ float sP[N_GRAPHS * HID];
    __shared__ float sC[N_GRAPHS];
    int t = threadIdx.x;
    for (int i = t; i < N_GRAPHS * HID; i += 256) sP[i] = 0.0f;
    if (t < N_GRAPHS) sC[t] = 0.0f;
    __syncthreads();

    int col = t & 63;
    int sub = t >> 6;  // 0..3
    int base = blockIdx.x * 256;
    for (int p = 0; p < 64; ++p) {
        int n = base + sub * 64 + p;
        if (n < N) {
            int g = batch[n] & (N_GRAPHS - 1);
            atomicAdd(&sP[g * HID + col], h[(long long)n * HID + col]);
            if (col == 0) atomicAdd(&sC[g], 1.0f);
        }
    }
    __syncthreads();
    for (int i = t; i < N_GRAPHS * HID; i += 256)
        if (sP[i] != 0.0f) atomicAdd(&pooled[i], sP[i]);
    if (t < N_GRAPHS && sC[t] != 0.0f) atomicAdd(&counts[t], sC[t]);
}

__global__ void head_kernel(const float* __restrict__ pooled, const float* __restrict__ counts,
                            const float* __restrict__ Wh, const float* __restrict__ bh,
                            float* __restrict__ out) {
    int g = threadIdx.x;
    if (g < N_GRAPHS) {
        float inv = 1.0f / fmaxf(counts[g], 1.0f);
        float acc = 0.0f;
        for (int c = 0; c < HID; ++c) acc += pooled[g * HID + c] * inv * Wh[c];
        out[g] = acc + bh[0];
    }
}

// ---------------- launch ----------------
extern "C" void kernel_launch(void* const* d_in, const int* in_sizes, int n_in,
                              void* d_out, int out_size, void* d_ws, size_t ws_size,
                              hipStream_t stream) {
    const float* x    = (const float*)d_in[0];
    const int*   eidx = (const int*)d_in[1];
    const int*   batch= (const int*)d_in[2];
    const float* Wl0 = (const float*)d_in[3];
    const float* bl0 = (const float*)d_in[4];
    const float* Wr0 = (const float*)d_in[5];
    const float* Wl1 = (const float*)d_in[6];
    const float* bl1 = (const float*)d_in[7];
    const float* Wr1 = (const float*)d_in[8];
    const float* Wl2 = (const float*)d_in[9];
    const float* bl2 = (const float*)d_in[10];
    const float* Wr2 = (const float*)d_in[11];
    const float* Wh  = (const float*)d_in[12];
    const float* bh  = (const float*)d_in[13];
    float* out = (float*)d_out;

    const int N = in_sizes[0] / 16;   // 250000
    const int E = in_sizes[1] / 2;    // 2000000
    const int ntiles = N / 16;        // 15625 (N divisible by 16)
    const int* src = eidx;
    const int* dst = eidx + E;

    // workspace carve-up (256B aligned)
    char* ws = (char*)d_ws;
    size_t off = 0;
    auto carve = [&](size_t bytes) -> float* {
        float* p = (float*)(ws + off);
        off = (off + bytes + 255) & ~(size_t)255;
        return p;
    };
    float* deg    = carve((size_t)N * 4);
    float* dinv   = carve((size_t)N * 4);
    float* msg    = carve((size_t)N * HID * 4);
    float* hA     = carve((size_t)N * HID * 4);
    float* hB     = carve((size_t)N * HID * 4);
    float* pooled = carve((size_t)N_GRAPHS * HID * 4);
    float* counts = carve((size_t)N_GRAPHS * 4);

    auto zgrid = [](long long n) -> int {
        long long b = (n + 255) / 256;
        return (int)(b > 4096 ? 4096 : b);
    };

    // degree + inverse
    zero_kernel<<<zgrid(N), 256, 0, stream>>>(deg, N);
    degree_kernel<<<(E + 255) / 256, 256, 0, stream>>>(dst, deg, E);
    deginv_kernel<<<(N + 255) / 256, 256, 0, stream>>>(deg, dinv, N);

    const int gemm_blocks = (ntiles + 3) / 4;

    // ---- layer 0: K = 16, ReLU ----
    zero_kernel<<<zgrid((long long)N * 16), 256, 0, stream>>>(msg, (long long)N * 16);
    {
        long long T = (long long)E * (16 / 4);
        scatter_add_kernel<16><<<(int)((T + 255) / 256), 256, 0, stream>>>(x, src, dst, msg, E);
    }
    sage_gemm_kernel<16, true><<<gemm_blocks, 128, 0, stream>>>(msg, dinv, x, Wl0, bl0, Wr0, hA, ntiles);

    // ---- layer 1: K = 64, ReLU ----
    zero_kernel<<<zgrid((long long)N * HID), 256, 0, stream>>>(msg, (long long)N * HID);
    {
        long long T = (long long)E * (HID / 4);
        scatter_add_kernel<64><<<(int)((T + 255) / 256), 256, 0, stream>>>(hA, src, dst, msg, E);
    }
    sage_gemm_kernel<64, true><<<gemm_blocks, 128, 0, stream>>>(msg, dinv, hA, Wl1, bl1, Wr1, hB, ntiles);

    // ---- layer 2: K = 64, no ReLU ----
    zero_kernel<<<zgrid((long long)N * HID), 256, 0, stream>>>(msg, (long long)N * HID);
    {
        long long T = (long long)E * (HID / 4);
        scatter_add_kernel<64><<<(int)((T + 255) / 256), 256, 0, stream>>>(hB, src, dst, msg, E);
    }
    sage_gemm_kernel<64, false><<<gemm_blocks, 128, 0, stream>>>(msg, dinv, hB, Wl2, bl2, Wr2, hA, ntiles);

    // ---- pool + head ----
    zero_kernel<<<1, 256, 0, stream>>>(pooled, N_GRAPHS * HID);
    zero_kernel<<<1, 64, 0, stream>>>(counts, N_GRAPHS);
    pool_kernel<<<(N + 255) / 256, 256, 0, stream>>>(hA, batch, pooled, counts, N);
    head_kernel<<<1, 64, 0, stream>>>(pooled, counts, Wh, bh, out);
}